// GAT_layer_58583353917588
// MI455X (gfx1250) — compile-verified
//
#include <hip/hip_runtime.h>
#include <hip/hip_bf16.h>

#define N_NODES 10000
#define DEG     32
#define D_IN    128
#define D_OUT   64
#define HEADS   4
#define E_EDGES (N_NODES * DEG)          // 320000
#define D_FULL  (HEADS * D_OUT)          // 256

typedef __attribute__((ext_vector_type(2))) float v2f;
typedef __attribute__((ext_vector_type(8))) float v8f;

// ---------------------------------------------------------------------------
// Kernel 1: edge extraction. One wave per row. b128 loads (512B/wave/iter)
// + 4 ballots per 128-column chunk; rank-via-popcount preserves the
// ascending-column order of jnp.nonzero. 400MB scan -> HBM bandwidth bound.
// ---------------------------------------------------------------------------
__global__ void gat_extract_edges(const float* __restrict__ A,
                                  int* __restrict__ col) {
    int wave = threadIdx.x >> 5;
    int lane = threadIdx.x & 31;
    int row  = blockIdx.x * 8 + wave;
    if (row >= N_NODES) return;
    const float* arow = A + (size_t)row * N_NODES;   // 40000B stride: 16B aligned
    int cnt = 0;

    const int VEC_END = (N_NODES / 128) * 128;       // 9984
    for (int base = 0; base < VEC_END; base += 128) {
        float4 v = *(const float4*)(arow + base + lane * 4);
        unsigned int lm = (v.x != 0.0f ? 1u : 0u) | (v.y != 0.0f ? 2u : 0u) |
                          (v.z != 0.0f ? 4u : 0u) | (v.w != 0.0f ? 8u : 0u);
        unsigned int m0 = (unsigned int)__ballot((lm >> 0) & 1u);
        unsigned int m1 = (unsigned int)__ballot((lm >> 1) & 1u);
        unsigned int m2 = (unsigned int)__ballot((lm >> 2) & 1u);
        unsigned int m3 = (unsigned int)__ballot((lm >> 3) & 1u);
        unsigned int below = (1u << lane) - 1u;
        // columns ordered by (lane, k): all lanes < lane precede this lane
        int pre = __popc(m0 & below) + __popc(m1 & below) +
                  __popc(m2 & below) + __popc(m3 & below);
        int own = 0;
        #pragma unroll
        for (int k = 0; k < 4; ++k) {
            if (lm & (1u << k)) {
                int slot = cnt + pre + own;
                if (slot < DEG) col[row * DEG + slot] = base + lane * 4 + k;
                ++own;
            }
        }
        cnt += __popc(m0) + __popc(m1) + __popc(m2) + __popc(m3);
    }
    // tail: remaining 16 columns, guarded scalar pass
    for (int base = VEC_END; base < N_NODES; base += 32) {
        int c = base + lane;
        float v = (c < N_NODES) ? arow[c] : 0.0f;
        unsigned int mask = (unsigned int)__ballot(v != 0.0f);
        if (v != 0.0f) {
            int rank = __popc(mask & ((1u << lane) - 1u));
            int slot = cnt + rank;
            if (slot < DEG) col[row * DEG + slot] = c;
        }
        cnt += __popc(mask);
    }
}

// ---------------------------------------------------------------------------
// Kernel 2: Xh = X @ W^T via V_WMMA_F32_16X16X4_F32 (fp32-exact matrix pipe).
// Block = 128 threads = 4 waves. Wave w computes rows [blockIdx*16,+16) x
// cols [w*64,+64): 4 accumulators, K-loop of 32 steps of 4.
// A 16x4 layout (ISA 7.12.2): v0 = K{0|2}, v1 = K{1|3} per lane half.
// B 4x16 layout mirrors A. D: VGPR r -> M = r + 8*laneHalf, N = lane&15.
// ---------------------------------------------------------------------------
__global__ void gat_gemm_xw(const float* __restrict__ X,
                            const float* __restrict__ W,
                            float* __restrict__ Xh) {
    int wave = threadIdx.x >> 5;
    int lane = threadIdx.x & 31;
    int m    = lane & 15;
    int half = lane >> 4;
    int row0 = blockIdx.x * 16;           // 625 blocks * 16 = 10000 exactly
    int col0 = wave * 64;                 // 4 waves cover 256 cols

    v8f acc0 = {}, acc1 = {}, acc2 = {}, acc3 = {};
    const float* xrow = X + (size_t)(row0 + m) * D_IN;

    #pragma unroll 4
    for (int kb = 0; kb < D_IN; kb += 4) {
        int ko = kb + 2 * half;
        v2f a;
        a.x = xrow[ko];
        a.y = xrow[ko + 1];

        v2f b0, b1, b2, b3;
        const float* wp = W + (size_t)(col0 + m) * D_IN + ko;
        b0.x = wp[0];             b0.y = wp[1];
        b1.x = wp[16 * D_IN];     b1.y = wp[16 * D_IN + 1];
        b2.x = wp[32 * D_IN];     b2.y = wp[32 * D_IN + 1];
        b3.x = wp[48 * D_IN];     b3.y = wp[48 * D_IN + 1];

        acc0 = __builtin_amdgcn_wmma_f32_16x16x4_f32(false, a, false, b0, (short)0, acc0, false, false);
        acc1 = __builtin_amdgcn_wmma_f32_16x16x4_f32(false, a, false, b1, (short)0, acc1, false, false);
        acc2 = __builtin_amdgcn_wmma_f32_16x16x4_f32(false, a, false, b2, (short)0, acc2, false, false);
        acc3 = __builtin_amdgcn_wmma_f32_16x16x4_f32(false, a, false, b3, (short)0, acc3, false, false);
    }

    #pragma unroll
    for (int r = 0; r < 8; ++r) {
        size_t o = (size_t)(row0 + r + 8 * half) * D_FULL + col0 + m;
        Xh[o]      = acc0[r];
        Xh[o + 16] = acc1[r];
        Xh[o + 32] = acc2[r];
        Xh[o + 48] = acc3[r];
    }
}

// ---------------------------------------------------------------------------
// Kernel 3: per-node per-head attention dots. Block per node, wave per head.
// ---------------------------------------------------------------------------
__global__ void gat_srcdst(const float* __restrict__ Xh,
                           const float* __restrict__ att,
                           float* __restrict__ src,
                           float* __restrict__ dst) {
    int n = blockIdx.x;
    int h = threadIdx.x >> 5;
    int l = threadIdx.x & 31;
    const float* x = Xh + (size_t)n * D_FULL + h * D_OUT;
    const float* a = att + h * (2 * D_OUT);
    float s = x[l] * a[l]      + x[l + 32] * a[l + 32];
    float d = x[l] * a[64 + l] + x[l + 32] * a[96 + l];
    #pragma unroll
    for (int off = 16; off; off >>= 1) {
        s += __shfl_xor(s, off, 32);
        d += __shfl_xor(d, off, 32);
    }
    if (l == 0) {
        src[n * HEADS + h] = s;
        dst[n * HEADS + h] = d;
    }
}

// ---------------------------------------------------------------------------
// Kernel 4: edge scores + per-block max (global softmax pass 1).
// E = 320000 = 1250 blocks * 256 threads exactly. dest i = e>>5.
// ---------------------------------------------------------------------------
__global__ void gat_scores_max(const float* __restrict__ src,
                               const float* __restrict__ dst,
                               const int* __restrict__ col,
                               float* __restrict__ scores,
                               float* __restrict__ blockmax) {
    __shared__ float red[256];
    int tid = threadIdx.x;
    int e = blockIdx.x * 256 + tid;
    int i = e >> 5;
    int j = col[e];
    float m[HEADS];
    #pragma unroll
    for (int h = 0; h < HEADS; ++h) {
        float s = fmaxf(src[i * HEADS + h] + dst[j * HEADS + h], 0.0f);
        scores[(size_t)e * HEADS + h] = s;
        m[h] = s;
    }
    for (int h = 0; h < HEADS; ++h) {
        red[tid] = m[h];
        __syncthreads();
        for (int s = 128; s; s >>= 1) {
            if (tid < s) red[tid] = fmaxf(red[tid], red[tid + s]);
            __syncthreads();
        }
        if (tid == 0) blockmax[blockIdx.x * HEADS + h] = red[0];
        __syncthreads();
    }
}

// ---------------------------------------------------------------------------
// Kernel 5: reduce block maxima -> gmax[4]. Deterministic tree reduction.
// ---------------------------------------------------------------------------
__global__ void gat_reduce_max(const float* __restrict__ blockmax, int nb,
                               float* __restrict__ gmax) {
    __shared__ float red[256];
    int tid = threadIdx.x;
    for (int h = 0; h < HEADS; ++h) {
        float m = -1e30f;
        for (int b = tid; b < nb; b += 256) m = fmaxf(m, blockmax[b * HEADS + h]);
        red[tid] = m;
        __syncthreads();
        for (int s = 128; s; s >>= 1) {
            if (tid < s) red[tid] = fmaxf(red[tid], red[tid + s]);
            __syncthreads();
        }
        if (tid == 0) gmax[h] = red[0];
        __syncthreads();
    }
}

// ---------------------------------------------------------------------------
// Kernel 6: exp(score - gmax) written in place + per-block sums.
// ---------------------------------------------------------------------------
__global__ void gat_expsum(float* __restrict__ scores,
                           const float* __restrict__ gmax,
                           float* __restrict__ blocksum) {
    __shared__ float red[256];
    int tid = threadIdx.x;
    int e = blockIdx.x * 256 + tid;
    float sm[HEADS];
    #pragma unroll
    for (int h = 0; h < HEADS; ++h) {
        float ex = expf(scores[(size_t)e * HEADS + h] - gmax[h]);
        scores[(size_t)e * HEADS + h] = ex;
        sm[h] = ex;
    }
    for (int h = 0; h < HEADS; ++h) {
        red[tid] = sm[h];
        __syncthreads();
        for (int s = 128; s; s >>= 1) {
            if (tid < s) red[tid] += red[tid + s];
            __syncthreads();
        }
        if (tid == 0) blocksum[blockIdx.x * HEADS + h] = red[0];
        __syncthreads();
    }
}

// ---------------------------------------------------------------------------
// Kernel 7: reduce block sums -> ginv[4] = 1/sum. Deterministic.
// ---------------------------------------------------------------------------
__global__ void gat_reduce_sum(const float* __restrict__ blocksum, int nb,
                               float* __restrict__ ginv) {
    __shared__ float red[256];
    int tid = threadIdx.x;
    for (int h = 0; h < HEADS; ++h) {
        float s = 0.0f;
        for (int b = tid; b < nb; b += 256) s += blocksum[b * HEADS + h];
        red[tid] = s;
        __syncthreads();
        for (int st = 128; st; st >>= 1) {
            if (tid < st) red[tid] += red[tid + st];
            __syncthreads();
        }
        if (tid == 0) ginv[h] = 1.0f / red[0];
        __syncthreads();
    }
}

// ---------------------------------------------------------------------------
// Kernel 8: aggregation. Block per node (256 threads = 256 output channels).
// Gathers are 1KB coalesced rows of Xh (L2-resident, 192MB L2 >> 10MB Xh).
// ---------------------------------------------------------------------------
__global__ void gat_aggregate(const int* __restrict__ col,
                              const float* __restrict__ scores,
                              const float* __restrict__ ginv,
                              const float* __restrict__ Xh,
                              float* __restrict__ out) {
    __shared__ float wsh[DEG * HEADS];   // 128 edge weights
    __shared__ int   jsh[DEG];
    int n = blockIdx.x;
    int tid = threadIdx.x;
    if (tid < DEG) jsh[tid] = col[n * DEG + tid];
    if (tid < DEG * HEADS) {
        int k = tid >> 2, h = tid & 3;
        wsh[tid] = scores[(size_t)(n * DEG + k) * HEADS + h] * ginv[h];
    }
    __syncthreads();
    int h = tid >> 6;                    // channel c = tid, head = c/64
    float acc = 0.0f;
    #pragma unroll 8
    for (int k = 0; k < DEG; ++k) {
        acc += wsh[k * HEADS + h] * Xh[(size_t)jsh[k] * D_FULL + tid];
    }
    out[(size_t)n * D_FULL + tid] = fmaxf(acc, 0.0f);
}

// ---------------------------------------------------------------------------
extern "C" void kernel_launch(void* const* d_in, const int* in_sizes, int n_in,
                              void* d_out, int out_size, void* d_ws, size_t ws_size,
                              hipStream_t stream) {
    const float* A   = (const float*)d_in[0];   // (1, N, N)
    const float* X   = (const float*)d_in[1];   // (1, N, 128)
    const float* W   = (const float*)d_in[2];   // (256, 128)
    const float* att = (const float*)d_in[3];   // (4, 128)
    float* out = (float*)d_out;                 // (1, N, 256)

    char* ws = (char*)d_ws;
    // workspace layout (bytes)
    int*   col      = (int*)  (ws + 0);           // E ints        = 1,280,000
    float* Xh       = (float*)(ws + 1280000);     // N*256 floats  = 10,240,000
    float* src      = (float*)(ws + 11520000);    // N*4 floats    = 160,000
    float* dst      = (float*)(ws + 11680000);    // N*4 floats    = 160,000
    float* scores   = (float*)(ws + 11840000);    // E*4 floats    = 5,120,000
    float* blockmax = (float*)(ws + 16960000);    // 1250*4 floats = 20,000
    float* blocksum = (float*)(ws + 16980000);    // 1250*4 floats = 20,000
    float* gmax     = (float*)(ws + 17000000);    // 4 floats
    float* ginv     = (float*)(ws + 17000064);    // 4 floats

    const int nb_e = E_EDGES / 256;               // 1250 edge blocks

    gat_extract_edges<<<N_NODES / 8, 256, 0, stream>>>(A, col);
    gat_gemm_xw      <<<N_NODES / 16, 128, 0, stream>>>(X, W, Xh);
    gat_srcdst       <<<N_NODES, 128, 0, stream>>>(Xh, att, src, dst);
    gat_scores_max   <<<nb_e, 256, 0, stream>>>(src, dst, col, scores, blockmax);
    gat_reduce_max   <<<1, 256, 0, stream>>>(blockmax, nb_e, gmax);
    gat_expsum       <<<nb_e, 256, 0, stream>>>(scores, gmax, blocksum);
    gat_reduce_sum   <<<1, 256, 0, stream>>>(blocksum, nb_e, ginv);
    gat_aggregate    <<<N_NODES, 256, 0, stream>>>(col, scores, ginv, Xh, out);
}